// CBOW_34119220199942
// MI455X (gfx1250) — compile-verified
//
#include <hip/hip_runtime.h>
#include <hip/hip_bf16.h>
#include <math.h>

// CBOW negative-sampling loss for MI455X (gfx1250, wave32).
// One wave per batch element; 16 dot products (1 context + 15 negatives vs
// emb_input) computed as a chain of 32 V_WMMA_F32_16X16X4_F32 accumulations.

typedef __attribute__((ext_vector_type(2))) float v2f;
typedef __attribute__((ext_vector_type(4))) float v4f;
typedef __attribute__((ext_vector_type(8))) float v8f;

#define EMB_D 128
#define NNEG 15
#define WAVES_PER_BLOCK 8
#define BLOCK_THREADS (WAVES_PER_BLOCK * 32)

__device__ __forceinline__ float logsig(float z) {
    // log(sigmoid(z)) = min(z,0) - log1p(exp(-|z|))   (numerically stable)
    return fminf(z, 0.0f) - log1pf(__expf(-fabsf(z)));
}

__global__ __launch_bounds__(BLOCK_THREADS)
void cbow_neg_loss_kernel(const int* __restrict__ target,
                          const int* __restrict__ context,
                          const int* __restrict__ neg_idx,
                          const float* __restrict__ dropout_mask,
                          const float* __restrict__ W_target,
                          const float* __restrict__ W_context,
                          float* __restrict__ block_partials)
{
    __shared__ float xbuf[WAVES_PER_BLOCK][EMB_D]; // emb_input per wave
    __shared__ float wsum[WAVES_PER_BLOCK];

    const int lane = threadIdx.x & 31;
    const int wid  = threadIdx.x >> 5;
    const int m    = lane & 15;   // A-matrix row owned by this lane
    const int hi   = lane >> 4;   // 0: K pair {4s,4s+1}, 1: K pair {4s+2,4s+3}
    const long long b = (long long)blockIdx.x * WAVES_PER_BLOCK + wid;

    // ---- stage emb_input = W_target[target[b]] * dropout_mask[b] into LDS
    const int tb = target[b];
    const float* xw = W_target + (long long)tb * EMB_D;
    const float* mk = dropout_mask + b * (long long)EMB_D;
    v4f xv = *(const v4f*)(xw + 4 * lane) * *(const v4f*)(mk + 4 * lane);
    *(v4f*)(&xbuf[wid][4 * lane]) = xv;  // same-wave LDS: in-order, no barrier

    // ---- per-lane A row: row 0 = context, rows 1..15 = negatives
    const int idx = (m == 0) ? context[b] : neg_idx[b * NNEG + (m - 1)];
    const v2f* rowA = (const v2f*)(W_context + (long long)idx * EMB_D);
    const v2f* xrow = (const v2f*)(&xbuf[wid][0]);

    // ---- D[16x16] += A[16x4] x B[4x16], 32 steps over K = 128.
    // B replicated across all 16 columns -> every column of D = score vector.
    v8f c = {};
#pragma unroll
    for (int s = 0; s < EMB_D / 4; ++s) {
        v2f a  = rowA[2 * s + hi];   // A: lane m -> row m, K = 4s + 2*hi
        v2f bb = xrow[2 * s + hi];   // B: broadcast from LDS, same K pattern
        c = __builtin_amdgcn_wmma_f32_16x16x4_f32(
                /*neg_a=*/false, a, /*neg_b=*/false, bb,
                /*c_mod=*/(short)0, c, /*reuse_a=*/false, /*reuse_b=*/false);
    }

    // D layout: VGPR r holds row r (lanes 0-15) / row r+8 (lanes 16-31).
    // Contribution of element b = sum over 16 rows of logsigmoid(-dot).
    float lsum = 0.0f;
#pragma unroll
    for (int r = 0; r < 8; ++r)
        lsum += logsig(-c[r]);
    const float loss_b = __shfl(lsum, 0, 32) + __shfl(lsum, 16, 32);

    if (lane == 0) wsum[wid] = loss_b;
    __syncthreads();
    if (threadIdx.x == 0) {
        float s = 0.0f;
#pragma unroll
        for (int w = 0; w < WAVES_PER_BLOCK; ++w) s += wsum[w];
        block_partials[blockIdx.x] = s;   // deterministic: no float atomics
    }
}

__global__ __launch_bounds__(256)
void reduce_partials_kernel(const float* __restrict__ partials, int n,
                            float scale, float* __restrict__ out)
{
    __shared__ float sm[256];
    float s = 0.0f;
    for (int i = threadIdx.x; i < n; i += 256) s += partials[i]; // fixed order
    sm[threadIdx.x] = s;
    __syncthreads();
    for (int off = 128; off > 0; off >>= 1) {
        if ((int)threadIdx.x < off) sm[threadIdx.x] += sm[threadIdx.x + off];
        __syncthreads();
    }
    if (threadIdx.x == 0) out[0] = sm[0] * scale;   // scale = -1/B
}

extern "C" void kernel_launch(void* const* d_in, const int* in_sizes, int n_in,
                              void* d_out, int out_size, void* d_ws, size_t ws_size,
                              hipStream_t stream)
{
    const int*   target       = (const int*)d_in[0];
    const int*   context      = (const int*)d_in[1];
    const int*   neg_idx      = (const int*)d_in[2];
    const float* dropout_mask = (const float*)d_in[3];
    const float* W_target     = (const float*)d_in[4];
    const float* W_context    = (const float*)d_in[5];
    float*       out          = (float*)d_out;

    const int B = in_sizes[0];                  // 131072 batch elements
    const int nblocks = B / WAVES_PER_BLOCK;    // one wave32 per element
    float* partials = (float*)d_ws;             // nblocks * 4 bytes scratch

    cbow_neg_loss_kernel<<<nblocks, BLOCK_THREADS, 0, stream>>>(
        target, context, neg_idx, dropout_mask, W_target, W_context, partials);

    reduce_partials_kernel<<<1, 256, 0, stream>>>(
        partials, nblocks, -1.0f / (float)B, out);
}